// ShearFlowPDELossNPINN_5119601017346
// MI455X (gfx1250) — compile-verified
//
#include <hip/hip_runtime.h>
#include <stdint.h>

// ---------------------------------------------------------------------------
// ShearFlow PINN PDE loss, MI455X (gfx1250).
// Bandwidth-bound stencil (176 MB in / 5 floats out, ~7.6 us HBM floor).
// One 256-thread block (8 wave32) per (batch, t_interior, y) row:
//   1) stage 24 rows (u/v/s x 7, p x 3; 48 KB) into LDS with CDNA5
//      GLOBAL_LOAD_ASYNC_TO_LDS_B64 (ASYNCcnt-tracked DMA, uniform SADDR,
//      lane byte-offset -> fully coalesced), s_wait_asynccnt 0 + barrier;
//   2) compute the upwind stencil entirely from LDS (no VMEM waits in body);
//   3) deterministic fp64 tree reduction (reusing the tile LDS) -> d_ws,
//      then a single-block fixed-order final reduce -> 5 fp32 losses.
// ---------------------------------------------------------------------------

namespace {
constexpr int WD = 512;                 // x extent (axis -1)
constexpr int HT = 256;                 // y extent (axis -2)
constexpr int NT = 21;                  // time slices
constexpr int TI = 19;                  // interior time slices
constexpr int NB = 4;                   // batch
constexpr int PLANE = HT * WD;          // 131072
constexpr int NBLOCKS = NB * TI * HT;   // 19456 row-blocks
constexpr int NROWS = 24;               // staged LDS rows per block

constexpr float DXc = 1.0f / 256.0f;    // 1/NX
constexpr float DYc = 2.0f / 512.0f;    // 2/NY (== 1/256)
constexpr float DTc = 0.1f;
constexpr float NUc = 0.0001f;
constexpr float DDc = 0.001f;
}

// Async DMA of one 512-float row into LDS: each of 256 lanes moves 8 bytes.
// GVS addressing: saddr = uniform row base (SGPR pair), vaddr = lane byte
// offset, vdst = LDS byte address (low 32 bits of the flat shared pointer
// == allocation-relative LDS offset, ISA sec 10.2 aperture mapping).
__device__ __forceinline__ void stage_row_async_b64(const float* __restrict__ grow,
                                                    float* lrow, int tid)
{
    const uint32_t loff = (uint32_t)(uintptr_t)(lrow + tid * 2);
    const uint32_t goff = (uint32_t)tid * 8u;
    asm volatile("global_load_async_to_lds_b64 %0, %1, %2"
                 :
                 : "v"(loff), "v"(goff), "s"(grow)
                 : "memory");
}

__device__ __forceinline__ float upwind_conv(
    float f,  float fxm, float fxm2, float fxp, float fxp2,
    float fym, float fym2, float fyp, float fyp2,
    float uce, float ucw, float vcn, float vcs)
{
    // jnp.roll(f, 1, -1) == f[x-1]; jnp.roll(f, -1, -1) == f[x+1] (same for y)
    float Fe = (uce >= 0.0f) ? (1.5f * f   - 0.5f * fxm)  : (1.5f * fxp - 0.5f * fxp2);
    float Fw = (ucw >= 0.0f) ? (1.5f * fxm - 0.5f * fxm2) : (1.5f * f   - 0.5f * fxp);
    float Fn = (vcn >= 0.0f) ? (1.5f * f   - 0.5f * fym)  : (1.5f * fyp - 0.5f * fyp2);
    float Fs = (vcs >= 0.0f) ? (1.5f * fym - 0.5f * fym2) : (1.5f * f   - 0.5f * fyp);
    return (uce * Fe - ucw * Fw) / DXc + (vcn * Fn - vcs * Fs) / DYc;
}

__global__ __launch_bounds__(256) void pde_row_kernel(
    const float* __restrict__ u, const float* __restrict__ v,
    const float* __restrict__ pr, const float* __restrict__ s,
    double* __restrict__ partial)
{
    // 48 KB tile; reused (after a barrier) as fp64 reduction scratch.
    __shared__ __align__(16) float tile[NROWS * WD];

    const int bid = blockIdx.x;
    const int y   = bid & (HT - 1);
    const int tmp = bid >> 8;           // / HT
    const int ti  = tmp % TI;
    const int b   = tmp / TI;
    const int tid = (int)threadIdx.x;

    const int base1 = b * (NT * PLANE) + (ti + 1) * PLANE;  // interior slice t = ti+1
    const int base0 = b * (NT * PLANE) +  ti      * PLANE;  // t   (for d/dt)
    const int base2 = b * (NT * PLANE) + (ti + 2) * PLANE;  // t+2

    const int ym  = (y - 1) & (HT - 1);
    const int ym2 = (y - 2) & (HT - 1);
    const int yp  = (y + 1) & (HT - 1);
    const int yp2 = (y + 2) & (HT - 1);

    // ---- Stage 24 rows via CDNA5 async-to-LDS DMA --------------------------
    // u rows 0..6: [y-2, y-1, y, y+1, y+2] @ t+1, [y] @ t, [y] @ t+2
    stage_row_async_b64(u + base1 + ym2 * WD, tile +  0 * WD, tid);
    stage_row_async_b64(u + base1 + ym  * WD, tile +  1 * WD, tid);
    stage_row_async_b64(u + base1 + y   * WD, tile +  2 * WD, tid);
    stage_row_async_b64(u + base1 + yp  * WD, tile +  3 * WD, tid);
    stage_row_async_b64(u + base1 + yp2 * WD, tile +  4 * WD, tid);
    stage_row_async_b64(u + base0 + y   * WD, tile +  5 * WD, tid);
    stage_row_async_b64(u + base2 + y   * WD, tile +  6 * WD, tid);
    // v rows 7..13
    stage_row_async_b64(v + base1 + ym2 * WD, tile +  7 * WD, tid);
    stage_row_async_b64(v + base1 + ym  * WD, tile +  8 * WD, tid);
    stage_row_async_b64(v + base1 + y   * WD, tile +  9 * WD, tid);
    stage_row_async_b64(v + base1 + yp  * WD, tile + 10 * WD, tid);
    stage_row_async_b64(v + base1 + yp2 * WD, tile + 11 * WD, tid);
    stage_row_async_b64(v + base0 + y   * WD, tile + 12 * WD, tid);
    stage_row_async_b64(v + base2 + y   * WD, tile + 13 * WD, tid);
    // s rows 14..20
    stage_row_async_b64(s + base1 + ym2 * WD, tile + 14 * WD, tid);
    stage_row_async_b64(s + base1 + ym  * WD, tile + 15 * WD, tid);
    stage_row_async_b64(s + base1 + y   * WD, tile + 16 * WD, tid);
    stage_row_async_b64(s + base1 + yp  * WD, tile + 17 * WD, tid);
    stage_row_async_b64(s + base1 + yp2 * WD, tile + 18 * WD, tid);
    stage_row_async_b64(s + base0 + y   * WD, tile + 19 * WD, tid);
    stage_row_async_b64(s + base2 + y   * WD, tile + 20 * WD, tid);
    // p rows 21..23: [y-1, y, y+1] @ t+1
    stage_row_async_b64(pr + base1 + ym * WD, tile + 21 * WD, tid);
    stage_row_async_b64(pr + base1 + y  * WD, tile + 22 * WD, tid);
    stage_row_async_b64(pr + base1 + yp * WD, tile + 23 * WD, tid);

    asm volatile("s_wait_asynccnt 0x0" ::: "memory");   // this wave's DMAs done
    __syncthreads();                                     // all waves' DMAs visible

    const float* U = tile;            // rows: ym2,ym,yc,yp,yp2,t0,t2
    const float* V = tile +  7 * WD;
    const float* S = tile + 14 * WD;
    const float* P = tile + 21 * WD;  // rows: ym,yc,yp

    double acc_div = 0.0, acc_u = 0.0, acc_v = 0.0, acc_s = 0.0;

    for (int x = tid; x < WD; x += 256) {
        const int xm  = (x - 1) & (WD - 1);
        const int xm2 = (x - 2) & (WD - 1);
        const int xp  = (x + 1) & (WD - 1);
        const int xp2 = (x + 2) & (WD - 1);

        const float uc   = U[2 * WD + x];
        const float uxm  = U[2 * WD + xm],  uxm2 = U[2 * WD + xm2];
        const float uxp  = U[2 * WD + xp],  uxp2 = U[2 * WD + xp2];
        const float uym  = U[1 * WD + x],   uym2 = U[0 * WD + x];
        const float uyp  = U[3 * WD + x],   uyp2 = U[4 * WD + x];
        const float ut0  = U[5 * WD + x],   ut2  = U[6 * WD + x];

        const float vc   = V[2 * WD + x];
        const float vxm  = V[2 * WD + xm],  vxm2 = V[2 * WD + xm2];
        const float vxp  = V[2 * WD + xp],  vxp2 = V[2 * WD + xp2];
        const float vym  = V[1 * WD + x],   vym2 = V[0 * WD + x];
        const float vyp  = V[3 * WD + x],   vyp2 = V[4 * WD + x];
        const float vt0  = V[5 * WD + x],   vt2  = V[6 * WD + x];

        const float sc   = S[2 * WD + x];
        const float sxm  = S[2 * WD + xm],  sxm2 = S[2 * WD + xm2];
        const float sxp  = S[2 * WD + xp],  sxp2 = S[2 * WD + xp2];
        const float sym  = S[1 * WD + x],   sym2 = S[0 * WD + x];
        const float syp  = S[3 * WD + x],   syp2 = S[4 * WD + x];
        const float st0  = S[5 * WD + x],   st2  = S[6 * WD + x];

        const float pxm = P[1 * WD + xm],  pxp = P[1 * WD + xp];
        const float pym = P[0 * WD + x],   pyp = P[2 * WD + x];

        // face velocities and divergence
        const float uce = 0.5f * (uc + uxp);
        const float ucw = 0.5f * (uc + uxm);
        const float vcn = 0.5f * (vc + vyp);
        const float vcs = 0.5f * (vc + vym);
        const float div = (uce - ucw) / DXc + (vcn - vcs) / DYc;

        const float conv_u = upwind_conv(uc, uxm, uxm2, uxp, uxp2,
                                         uym, uym2, uyp, uyp2, uce, ucw, vcn, vcs);
        const float conv_v = upwind_conv(vc, vxm, vxm2, vxp, vxp2,
                                         vym, vym2, vyp, vyp2, uce, ucw, vcn, vcs);
        const float conv_s = upwind_conv(sc, sxm, sxm2, sxp, sxp2,
                                         sym, sym2, syp, syp2, uce, ucw, vcn, vcs);

        const float lap_u = (uxp - 2.0f * uc + uxm) / (DXc * DXc)
                          + (uyp - 2.0f * uc + uym) / (DYc * DYc);
        const float lap_v = (vxp - 2.0f * vc + vxm) / (DXc * DXc)
                          + (vyp - 2.0f * vc + vym) / (DYc * DYc);
        const float lap_s = (sxp - 2.0f * sc + sxm) / (DXc * DXc)
                          + (syp - 2.0f * sc + sym) / (DYc * DYc);

        const float Ru = (ut2 - ut0) / (2.0f * DTc) + conv_u
                       + (pxp - pxm) / (2.0f * DXc) - NUc * lap_u - uc * div;
        const float Rv = (vt2 - vt0) / (2.0f * DTc) + conv_v
                       + (pyp - pym) / (2.0f * DYc) - NUc * lap_v - vc * div;
        const float Rs = (st2 - st0) / (2.0f * DTc) + conv_s
                       - DDc * lap_s - sc * div;

        acc_div += (double)div * (double)div;
        acc_u   += (double)Ru  * (double)Ru;
        acc_v   += (double)Rv  * (double)Rv;
        acc_s   += (double)Rs  * (double)Rs;
    }

    // ---- Deterministic block tree reduction (reuse tile LDS as fp64) -------
    __syncthreads();                    // everyone done reading the tile
    double* sm = (double*)tile;         // [4][256] doubles = 8 KB < 48 KB
    sm[0 * 256 + tid] = acc_div;
    sm[1 * 256 + tid] = acc_u;
    sm[2 * 256 + tid] = acc_v;
    sm[3 * 256 + tid] = acc_s;
    __syncthreads();
    for (int off = 128; off > 0; off >>= 1) {
        if (tid < off) {
            sm[0 * 256 + tid] += sm[0 * 256 + tid + off];
            sm[1 * 256 + tid] += sm[1 * 256 + tid + off];
            sm[2 * 256 + tid] += sm[2 * 256 + tid + off];
            sm[3 * 256 + tid] += sm[3 * 256 + tid + off];
        }
        __syncthreads();
    }
    if (tid == 0) {
        double* dst = partial + (size_t)bid * 4;
        dst[0] = sm[0 * 256]; dst[1] = sm[1 * 256];
        dst[2] = sm[2 * 256]; dst[3] = sm[3 * 256];
    }
}

__global__ __launch_bounds__(256) void pde_final_reduce(
    const double* __restrict__ partial, float* __restrict__ out)
{
    __shared__ double sm[4][256];
    const int tid = (int)threadIdx.x;
    double a0 = 0.0, a1 = 0.0, a2 = 0.0, a3 = 0.0;
    for (int i = tid; i < NBLOCKS; i += 256) {
        const double* pi = partial + (size_t)i * 4;
        a0 += pi[0]; a1 += pi[1]; a2 += pi[2]; a3 += pi[3];
    }
    sm[0][tid] = a0; sm[1][tid] = a1; sm[2][tid] = a2; sm[3][tid] = a3;
    __syncthreads();
    for (int off = 128; off > 0; off >>= 1) {
        if (tid < off) {
            sm[0][tid] += sm[0][tid + off];
            sm[1][tid] += sm[1][tid + off];
            sm[2][tid] += sm[2][tid + off];
            sm[3][tid] += sm[3][tid + off];
        }
        __syncthreads();
    }
    if (tid == 0) {
        const double invN = 1.0 / ((double)NB * TI * HT * WD);   // 1 / 9961472
        const double mdiv = sm[0][0] * invN;
        const double mu   = sm[1][0] * invN;
        const double mv   = sm[2][0] * invN;
        const double ms   = sm[3][0] * invN;
        out[0] = (float)(mdiv + mu + mv + ms);  // total
        out[1] = (float)mdiv;                   // loss_cont
        out[2] = (float)mu;                     // loss_u
        out[3] = (float)mv;                     // loss_v
        out[4] = (float)ms;                     // loss_s
    }
}

extern "C" void kernel_launch(void* const* d_in, const int* in_sizes, int n_in,
                              void* d_out, int out_size, void* d_ws, size_t ws_size,
                              hipStream_t stream)
{
    (void)in_sizes; (void)n_in; (void)out_size; (void)ws_size;
    const float* u  = (const float*)d_in[0];
    const float* v  = (const float*)d_in[1];
    const float* pr = (const float*)d_in[2];
    const float* s  = (const float*)d_in[3];
    double* partial = (double*)d_ws;            // NBLOCKS*4 doubles = 608 KB
    float*  out     = (float*)d_out;

    pde_row_kernel<<<NBLOCKS, 256, 0, stream>>>(u, v, pr, s, partial);
    pde_final_reduce<<<1, 256, 0, stream>>>(partial, out);
}